// tgt_MultiHeadedAttention_34711925686816
// MI455X (gfx1250) — compile-verified
//
#include <hip/hip_runtime.h>
#include <hip/hip_bf16.h>

// Single-token MHA, algebraically folded:
//   scores[h,n] = qk[h,:] . key[n,:] + c[h]        (qk = Wk_h^T q_h, c = q_h.bk_h)
//   pv[h,:]     = sum_n p[h,n] * value[n,:]
//   x[h,:]      = Wv_h @ pv[h,:] + bv_h            (softmax sums to 1)
// Memory-bound: ~144 MB streamed once; both big GEMMs (M=16) run on
// V_WMMA_F32_16X16X4_F32 with split-K for occupancy (2048 waves each).

typedef __attribute__((ext_vector_type(2))) float v2f;
typedef __attribute__((ext_vector_type(8))) float v8f;

#define D_MODEL 1024
#define NWORDS  16384
#define NHEADS  16
#define DK      64

// ---------------- small projection kernels (f32 VALU, trivial cost) ---------

// q[o] = bq[o] + query . Wq[o,:]   (o = 0..1023)  -- grid 4x256
__global__ void k_proj_q(const float* __restrict__ query,
                         const float* __restrict__ Wq,
                         const float* __restrict__ bq,
                         float* __restrict__ q) {
  int o = blockIdx.x * blockDim.x + threadIdx.x;
  const float* w = Wq + (size_t)o * D_MODEL;
  float acc = bq[o];
  #pragma unroll 4
  for (int d = 0; d < D_MODEL; ++d) acc = fmaf(query[d], w[d], acc);
  q[o] = acc;
}

// qk[h*1024+d] = sum_j q[h*64+j] * Wk[h*64+j, d];  c[h] = sum_j q[h*64+j]*bk[h*64+j]
// grid 64x256 (16384 outputs, 64-MAC dot each; coalesced over d)
__global__ void k_fold_qk(const float* __restrict__ q,
                          const float* __restrict__ Wk,
                          const float* __restrict__ bk,
                          float* __restrict__ qk,
                          float* __restrict__ c) {
  int o = blockIdx.x * blockDim.x + threadIdx.x;      // 0..16383
  int h = o >> 10, d = o & (D_MODEL - 1);
  const float* qh = q + h * DK;
  const float* w  = Wk + ((size_t)h * DK) * D_MODEL + d;
  float acc = 0.f;
  #pragma unroll 4
  for (int j = 0; j < DK; ++j) acc = fmaf(qh[j], w[(size_t)j * D_MODEL], acc);
  qk[o] = acc;
  if (blockIdx.x == 0 && threadIdx.x < NHEADS) {
    int hh = threadIdx.x;
    float cc = 0.f;
    for (int j = 0; j < DK; ++j) cc = fmaf(q[hh * DK + j], bk[hh * DK + j], cc);
    c[hh] = cc;
  }
}

// ---------------- scores GEMM: [16 x 1024] @ [1024 x 16384] via f32 WMMA ----
// A = qk (M=16 heads, K=d), B = key^T (B[k][n] = key[n][k]).
// Split-K over d: 2 chunks of 512. 1024 n-tiles * 2 chunks = 2048 waves.
// Partial tiles -> scpart[chunk][h][n].
__global__ void k_scores_wmma(const float* __restrict__ key,
                              const float* __restrict__ qk,
                              float* __restrict__ scpart) {
  const int tid  = threadIdx.x;
  const int lane = tid & 31;
  const int wgid = blockIdx.x * 8 + (tid >> 5);   // 0..2047
  const int tileN = wgid & 1023;
  const int chunk = wgid >> 10;                   // 0..1
  const int n0 = tileN * 16;
  const int d0 = chunk * 512;
  const int r  = lane & 15;
  const int hi = lane >> 4;

  // A frag: lane holds qk[r][k + 2*hi + {0,1}]  (contiguous -> float2)
  const float* aptr = qk + (size_t)r * D_MODEL + d0 + 2 * hi;
  // B frag: B[k][n] = key[n][k]; lane holds key[n0+r][k + 2*hi + {0,1}] (float2)
  const float* bptr = key + (size_t)(n0 + r) * D_MODEL + d0 + 2 * hi;

  v8f acc = {};
  #pragma unroll 4
  for (int k = 0; k < 512; k += 4) {
    v2f a = *(const v2f*)(aptr + k);
    v2f b = *(const v2f*)(bptr + k);
    acc = __builtin_amdgcn_wmma_f32_16x16x4_f32(false, a, false, b,
                                                (short)0, acc, false, false);
  }
  float* out = scpart + (size_t)chunk * (NHEADS * NWORDS);
  #pragma unroll
  for (int g = 0; g < 8; ++g) {
    int h = g + 8 * hi;                           // C/D layout: M = g + 8*(lane>>4)
    out[(size_t)h * NWORDS + n0 + r] = acc[g];
  }
}

// ---------------- softmax over n per head (adds the two d-chunks + c[h]) ----
__global__ void k_softmax(const float* __restrict__ scpart,
                          const float* __restrict__ c,
                          float* __restrict__ probs) {
  __shared__ float red[256];
  const int h = blockIdx.x, tid = threadIdx.x;
  const float bias = c[h];
  const float* s0 = scpart + (size_t)h * NWORDS;
  const float* s1 = scpart + (size_t)NHEADS * NWORDS + (size_t)h * NWORDS;
  float mx = -3.0e38f;
  for (int n = tid; n < NWORDS; n += 256) mx = fmaxf(mx, s0[n] + s1[n] + bias);
  red[tid] = mx; __syncthreads();
  for (int s = 128; s > 0; s >>= 1) {
    if (tid < s) red[tid] = fmaxf(red[tid], red[tid + s]);
    __syncthreads();
  }
  mx = red[0]; __syncthreads();
  float sum = 0.f;
  float* p = probs + (size_t)h * NWORDS;
  for (int n = tid; n < NWORDS; n += 256) {
    float e = __expf(s0[n] + s1[n] + bias - mx);
    p[n] = e; sum += e;
  }
  red[tid] = sum; __syncthreads();
  for (int s = 128; s > 0; s >>= 1) {
    if (tid < s) red[tid] += red[tid + s];
    __syncthreads();
  }
  float inv = 1.0f / red[0];
  for (int n = tid; n < NWORDS; n += 256) p[n] *= inv;
}

// ---------------- pv GEMM: [16 x 16384] @ [16384 x 1024] via f32 WMMA -------
// A = probs (M=16, K=n), B = value (row-major [K x N], N=d).
// Split-K over tokens: 32 chunks of 512. 64 d-tiles * 32 = 2048 waves.
__global__ void k_pv_wmma(const float* __restrict__ value,
                          const float* __restrict__ probs,
                          float* __restrict__ pvpart) {
  const int tid  = threadIdx.x;
  const int lane = tid & 31;
  const int wgid = blockIdx.x * 8 + (tid >> 5);   // 0..2047
  const int tileD = wgid & 63;
  const int chunk = wgid >> 6;                    // 0..31
  const int d0 = tileD * 16;
  const int k0 = chunk * 512;
  const int r  = lane & 15;
  const int hi = lane >> 4;

  const float* aptr = probs + (size_t)r * NWORDS + k0 + 2 * hi;          // float2
  const float* bptr = value + (size_t)(k0 + 2 * hi) * D_MODEL + d0 + r;  // 2x b32

  v8f acc = {};
  #pragma unroll 4
  for (int k = 0; k < 512; k += 4) {
    v2f a = *(const v2f*)(aptr + k);
    v2f b;
    b.x = bptr[(size_t)k * D_MODEL];              // B[k + 2*hi + 0][d0+r]
    b.y = bptr[(size_t)(k + 1) * D_MODEL];        // B[k + 2*hi + 1][d0+r]
    acc = __builtin_amdgcn_wmma_f32_16x16x4_f32(false, a, false, b,
                                                (short)0, acc, false, false);
  }
  float* out = pvpart + (size_t)chunk * (NHEADS * D_MODEL);
  #pragma unroll
  for (int g = 0; g < 8; ++g) {
    int h = g + 8 * hi;
    out[(size_t)h * D_MODEL + d0 + r] = acc[g];
  }
}

// ---------------- epilogue kernels ------------------------------------------

// pv[i] = sum over 32 split-K partials (deterministic order). grid 64x256.
__global__ void k_reduce_pv(const float* __restrict__ pvpart, float* __restrict__ pv) {
  int i = blockIdx.x * blockDim.x + threadIdx.x;  // 0..16383
  float acc = 0.f;
  #pragma unroll
  for (int ch = 0; ch < 32; ++ch) acc += pvpart[(size_t)ch * (NHEADS * D_MODEL) + i];
  pv[i] = acc;
}

// x[h*64+dp] = bv[o] + pv[h,:] . Wv[o,:]. grid 4x256.
__global__ void k_proj_x(const float* __restrict__ pv, const float* __restrict__ Wv,
                         const float* __restrict__ bv, float* __restrict__ x) {
  int o = blockIdx.x * blockDim.x + threadIdx.x;  // 0..1023
  int h = o >> 6;
  const float* pvh = pv + (size_t)h * D_MODEL;
  const float* w = Wv + (size_t)o * D_MODEL;
  float acc = bv[o];
  #pragma unroll 4
  for (int d = 0; d < D_MODEL; ++d) acc = fmaf(pvh[d], w[d], acc);
  x[o] = acc;
}

// out[i] = bo[i] + x . Wo[i,:]. grid 4x256.
__global__ void k_proj_o(const float* __restrict__ x, const float* __restrict__ Wo,
                         const float* __restrict__ bo, float* __restrict__ out) {
  int i = blockIdx.x * blockDim.x + threadIdx.x;  // 0..1023
  const float* w = Wo + (size_t)i * D_MODEL;
  float acc = bo[i];
  #pragma unroll 4
  for (int j = 0; j < D_MODEL; ++j) acc = fmaf(x[j], w[j], acc);
  out[i] = acc;
}

// ---------------- host launcher ---------------------------------------------

extern "C" void kernel_launch(void* const* d_in, const int* in_sizes, int n_in,
                              void* d_out, int out_size, void* d_ws, size_t ws_size,
                              hipStream_t stream) {
  const float* query = (const float*)d_in[0];
  const float* key   = (const float*)d_in[1];
  const float* value = (const float*)d_in[2];
  const float* Wq    = (const float*)d_in[3];
  const float* bq    = (const float*)d_in[4];
  const float* Wk    = (const float*)d_in[5];
  const float* bk    = (const float*)d_in[6];
  const float* Wv    = (const float*)d_in[7];
  const float* bv    = (const float*)d_in[8];
  const float* Wo    = (const float*)d_in[9];
  const float* bo    = (const float*)d_in[10];
  float* out = (float*)d_out;

  // workspace layout (floats); total ~5.2 MB
  float* ws = (float*)d_ws;
  float* q      = ws;                        // 1024
  float* qk     = q + 1024;                  // 16*1024
  float* c      = qk + NHEADS * D_MODEL;     // 16
  float* scpart = c + 16;                    // 2 * 16 * 16384
  float* probs  = scpart + 2 * NHEADS * NWORDS;   // 16 * 16384
  float* pvpart = probs + NHEADS * NWORDS;   // 32 * 16 * 1024
  float* pv     = pvpart + 32 * NHEADS * D_MODEL; // 16 * 1024
  float* x      = pv + NHEADS * D_MODEL;     // 1024

  k_proj_q    <<<   4, 256, 0, stream>>>(query, Wq, bq, q);
  k_fold_qk   <<<  64, 256, 0, stream>>>(q, Wk, bk, qk, c);
  k_scores_wmma<<< 256, 256, 0, stream>>>(key, qk, scpart);
  k_softmax   <<<  16, 256, 0, stream>>>(scpart, c, probs);
  k_pv_wmma   <<< 256, 256, 0, stream>>>(value, probs, pvpart);
  k_reduce_pv <<<  64, 256, 0, stream>>>(pvpart, pv);
  k_proj_x    <<<   4, 256, 0, stream>>>(pv, Wv, bv, x);
  k_proj_o    <<<   4, 256, 0, stream>>>(x, Wo, bo, out);
}